// TrainModel_18528488914975
// MI455X (gfx1250) — compile-verified
//
#include <hip/hip_runtime.h>

typedef __attribute__((ext_vector_type(2))) float v2f;
typedef __attribute__((ext_vector_type(8))) float v8f;

// ---------------------------------------------------------------------------
// utility: fill buffer with constant
// ---------------------------------------------------------------------------
__global__ void fill_f32(float* __restrict__ p, float v, long n) {
  long i = (long)blockIdx.x * blockDim.x + threadIdx.x;
  long stride = (long)gridDim.x * blockDim.x;
  for (; i < n; i += stride) p[i] = v;
}

// ---------------------------------------------------------------------------
// degree accumulation: deg[col[e]] += ew[e]   (deg pre-filled with 1.0 = self loop)
// ---------------------------------------------------------------------------
__global__ void deg_accum(float* __restrict__ deg, const long long* __restrict__ col,
                          const float* __restrict__ ew, int E) {
  int e = blockIdx.x * blockDim.x + threadIdx.x;
  if (e < E) atomicAdd(&deg[col[e]], ew[e]);
}

__global__ void dinv_kernel(const float* __restrict__ deg, float* __restrict__ dinv, int N) {
  int i = blockIdx.x * blockDim.x + threadIdx.x;
  if (i < N) {
    float d = deg[i];
    dinv[i] = (d > 0.f) ? rsqrtf(d) : 0.f;
  }
}

// ---------------------------------------------------------------------------
// Dense GEMM  C[M,F] = A[M,K] x B[K,F]  (f32, row major), K <= 128.
//
// Block = 256 threads = 8 waves. Block tile: 128(M) x 64(N).
//  * The B panel B[0..K-1][nBlk..nBlk+63] (<=32 KB) is cooperatively staged
//    into LDS once per block, then served to all 8 waves (ds_load).
//  * Each wave computes a 16(M) x 64(N) strip with FOUR v8f accumulators,
//    so one A operand (v2f) feeds 4 consecutive v_wmma_f32_16x16x4_f32.
//
// f32 16x4 A operand layout (ISA 7.12.2): lanes 0-15 hold M=lane, K={k,k+1};
// lanes 16-31 hold M=lane-16, K={k+2,k+3}. B (4x16) symmetric. D: VGPR v
// holds row M = v + 8*(lane>=16), col N = lane&15.
// ---------------------------------------------------------------------------
__global__ __launch_bounds__(256) void gemm_f32_wmma(const float* __restrict__ A,
                                                     const float* __restrict__ B,
                                                     float* __restrict__ C,
                                                     int M, int K, int F) {
  __shared__ float Bs[128 * 64];                 // 32 KB B panel

  const int tid   = threadIdx.x;
  const int lane  = tid & 31;
  const int wv    = tid >> 5;
  const int nBlk  = blockIdx.y * 64;

  // ---- stage B panel into LDS (coalesced: 64 floats per row) ----
  for (int idx = tid; idx < K * 64; idx += 256) {
    int kk = idx >> 6;
    int nn = idx & 63;
    Bs[idx] = B[(long)kk * F + nBlk + nn];
  }
  __syncthreads();

  const int mBase = blockIdx.x * 128 + wv * 16;
  if (mBase >= M) return;                         // wave-uniform: EXEC stays full

  const int l15     = lane & 15;
  const int halfSel = lane >> 4;                  // 0: lanes 0-15, 1: lanes 16-31
  const int kOff    = halfSel * 2;                // K sub-offset within group of 4
  int mRow = mBase + l15;
  if (mRow >= M) mRow = M - 1;                    // clamp: feeds only unstored D rows

  const float* __restrict__ arow = A + (long)mRow * K;

  v8f acc[4] = {};
  for (int k = 0; k < K; k += 4) {
    v2f a;
    a.x = arow[k + kOff];
    a.y = arow[k + kOff + 1];
    const float* b0p = &Bs[(k + kOff)     * 64];
    const float* b1p = &Bs[(k + kOff + 1) * 64];
#pragma unroll
    for (int t = 0; t < 4; ++t) {
      v2f b;
      b.x = b0p[t * 16 + l15];
      b.y = b1p[t * 16 + l15];
      acc[t] = __builtin_amdgcn_wmma_f32_16x16x4_f32(
          /*neg_a=*/false, a, /*neg_b=*/false, b,
          /*c_mod=*/(short)0, acc[t], /*reuse_a=*/false, /*reuse_b=*/false);
    }
  }

  const int mOff = halfSel * 8;
#pragma unroll
  for (int t = 0; t < 4; ++t) {
    int n = nBlk + t * 16 + l15;
#pragma unroll
    for (int v = 0; v < 8; ++v) {
      int m = mBase + mOff + v;
      if (m < M) C[(long)m * F + n] = acc[t][v];
    }
  }
}

// ---------------------------------------------------------------------------
// Edge aggregation: agg[col[e], :] += feat[row[e], :] * (dinv[row]*ew*dinv[col])
// One wave per edge; lanes stride the feature dimension (coalesced).
// global_prefetch of the gathered row hides the index->row pointer chase.
// ---------------------------------------------------------------------------
__global__ void edge_aggregate(const float* __restrict__ feat,
                               const long long* __restrict__ row,
                               const long long* __restrict__ col,
                               const float* __restrict__ ew,
                               const float* __restrict__ dinv,
                               float* __restrict__ agg, int E, int F) {
  const int epb = blockDim.x >> 5;               // edges per block
  const int e   = blockIdx.x * epb + (threadIdx.x >> 5);
  if (e >= E) return;
  const int lane = threadIdx.x & 31;
  const long long r = row[e];
  const long long c = col[e];
  const float* __restrict__ src = feat + (long)r * F;
  __builtin_prefetch(src + lane * 2, 0, 3);      // global_prefetch_b8 (row is L2-hot)
  const float norm = dinv[r] * ew[e] * dinv[c];
  float* __restrict__ dst = agg + (long)c * F;
  for (int j = lane; j < F; j += 32) {
    atomicAdd(&dst[j], src[j] * norm);           // global_atomic_add_f32
  }
}

// ---------------------------------------------------------------------------
// Epilogue: out = relu(agg + feat*dinv^2 (self loop) + bias)
// ---------------------------------------------------------------------------
__global__ void epilogue(const float* __restrict__ agg, const float* __restrict__ feat,
                         const float* __restrict__ dinv, const float* __restrict__ bias,
                         float* __restrict__ out, int N, int F) {
  long i = (long)blockIdx.x * blockDim.x + threadIdx.x;
  long total = (long)N * F;
  if (i >= total) return;
  int node = (int)(i / F);
  int f    = (int)(i % F);
  float di = dinv[node];
  float v  = agg[i] + feat[i] * di * di + bias[f];
  out[i] = v > 0.f ? v : 0.f;
}

// ---------------------------------------------------------------------------
extern "C" void kernel_launch(void* const* d_in, const int* in_sizes, int n_in,
                              void* d_out, int out_size, void* d_ws, size_t ws_size,
                              hipStream_t stream) {
  const float*     x    = (const float*)d_in[0];
  const long long* eidx = (const long long*)d_in[1];   // [2,E] int64
  const float*     ew   = (const float*)d_in[2];
  const float*     W0   = (const float*)d_in[3];
  const float*     b0   = (const float*)d_in[4];
  const float*     W1   = (const float*)d_in[5];
  const float*     b1   = (const float*)d_in[6];

  const int Fin  = 128;
  const int Fhid = in_sizes[4];          // 128 (len of b0)
  const int Fout = in_sizes[6];          // 64  (len of b1)
  const int E    = in_sizes[2];          // 800000
  const int N    = in_sizes[0] / Fin;    // 50000

  const long long* rowp = eidx;          // sources j
  const long long* colp = eidx + E;      // targets i

  // workspace carving (floats)
  float* ws   = (float*)d_ws;
  float* deg  = ws;                         // N
  float* dinv = deg + N;                    // N
  float* xw   = dinv + N;                   // N*Fhid  (reused for layer 2, N*Fout)
  float* agg  = xw + (long)N * Fhid;        // N*Fhid  (reused for layer 2)
  float* h1   = agg + (long)N * Fhid;       // N*Fhid
  float* out  = (float*)d_out;              // N*Fout

  // ---- normalization coefficients ----
  fill_f32<<<512, 256, 0, stream>>>(deg, 1.0f, N);                 // self-loop weight 1
  deg_accum<<<(E + 255) / 256, 256, 0, stream>>>(deg, colp, ew, E);
  dinv_kernel<<<(N + 255) / 256, 256, 0, stream>>>(deg, dinv, N);

  const int mBlocks = (N + 127) / 128;

  // ---- layer 1: h1 = relu( Â (x W0) + b0 ) ----
  fill_f32<<<2048, 256, 0, stream>>>(agg, 0.f, (long)N * Fhid);
  gemm_f32_wmma<<<dim3(mBlocks, (Fhid + 63) / 64), 256, 0, stream>>>(x, W0, xw, N, Fin, Fhid);
  edge_aggregate<<<(E + 7) / 8, 256, 0, stream>>>(xw, rowp, colp, ew, dinv, agg, E, Fhid);
  epilogue<<<(int)(((long)N * Fhid + 255) / 256), 256, 0, stream>>>(agg, xw, dinv, b0, h1, N, Fhid);

  // ---- layer 2: out = relu( Â (h1 W1) + b1 ) ----
  fill_f32<<<2048, 256, 0, stream>>>(agg, 0.f, (long)N * Fout);
  gemm_f32_wmma<<<dim3(mBlocks, (Fout + 63) / 64), 256, 0, stream>>>(h1, W1, xw, N, Fhid, Fout);
  edge_aggregate<<<(E + 7) / 8, 256, 0, stream>>>(xw, rowp, colp, ew, dinv, agg, E, Fout);
  epilogue<<<(int)(((long)N * Fout + 255) / 256), 256, 0, stream>>>(agg, xw, dinv, b1, out, N, Fout);
}